// IWeightedSymmetricTensorProduct_52879637348695
// MI455X (gfx1250) — compile-verified
//
#include <hip/hip_runtime.h>
#include <stdint.h>

#define ZROWS 20000
#define NB    50000
#define SDIM  16
#define UDIM  32
#define NP    128
#define ROW   (SDIM * UDIM)   // 512 floats per row

// Workspace layout (uint32 words):
//  [0 .. 48]    bucket offsets; bucket key = s2*3 + (deg-1), 48 buckets + end
//  [64 .. 831]  384 path records, 2 words each: { p0|s1a<<4|s1b<<8|s1c<<12 , coef }
#define WS_OFF  0
#define WS_REC  64

// ---------------------------------------------------------------------------
// Setup: counting-sort the 384 paths by (p2, degree) into a compact program.
// Single thread; ~1k iterations; runs every launch (deterministic).
// ---------------------------------------------------------------------------
__global__ void setup_paths(const float* c1, const float* c2, const float* c3,
                            const int* p0_1, const int* p1_1, const int* p2_1,
                            const int* p0_2, const int* p1_2, const int* p2_2,
                            const int* p0_3, const int* p1_3, const int* p2_3,
                            unsigned* ws) {
  if (threadIdx.x != 0 || blockIdx.x != 0) return;
  unsigned counts[48];
#pragma unroll 1
  for (int k = 0; k < 48; ++k) counts[k] = 0;
#pragma unroll 1
  for (int p = 0; p < NP; ++p) counts[p2_1[p] * 3 + 0]++;
#pragma unroll 1
  for (int p = 0; p < NP; ++p) counts[p2_2[p] * 3 + 1]++;
#pragma unroll 1
  for (int p = 0; p < NP; ++p) counts[p2_3[p] * 3 + 2]++;
  unsigned run = 0;
#pragma unroll 1
  for (int k = 0; k < 48; ++k) {
    unsigned c = counts[k];
    ws[WS_OFF + k] = run;
    counts[k] = run;          // reuse as write cursor
    run += c;
  }
  ws[WS_OFF + 48] = run;      // == 384
#pragma unroll 1
  for (int p = 0; p < NP; ++p) {
    unsigned i = counts[p2_1[p] * 3 + 0]++;
    ws[WS_REC + 2 * i]     = (unsigned)p0_1[p] | ((unsigned)p1_1[p] << 4);
    ((float*)ws)[WS_REC + 2 * i + 1] = c1[p];
  }
#pragma unroll 1
  for (int p = 0; p < NP; ++p) {
    unsigned i = counts[p2_2[p] * 3 + 1]++;
    ws[WS_REC + 2 * i] = (unsigned)p0_2[p] | ((unsigned)p1_2[2 * p] << 4) |
                         ((unsigned)p1_2[2 * p + 1] << 8);
    ((float*)ws)[WS_REC + 2 * i + 1] = c2[p];
  }
#pragma unroll 1
  for (int p = 0; p < NP; ++p) {
    unsigned i = counts[p2_3[p] * 3 + 2]++;
    ws[WS_REC + 2 * i] = (unsigned)p0_3[p] | ((unsigned)p1_3[3 * p] << 4) |
                         ((unsigned)p1_3[3 * p + 1] << 8) |
                         ((unsigned)p1_3[3 * p + 2] << 12);
    ((float*)ws)[WS_REC + 2 * i + 1] = c3[p];
  }
}

// ---------------------------------------------------------------------------
// Main kernel: one wave per PAIR of batch rows, lane == u channel (wave32==U).
// LDS layout per buffer (floats): X0[512] X1[512] G0[512] G1[512]; row pairs
// sit 2048B apart so each gather of both rows merges into ds_read2st64_b32.
// Double-buffered async global->LDS staging (ASYNCcnt completes in order:
// wait <=16 releases the current buffer while the next pair is in flight).
// Paths sorted by output slot p2 -> accumulators stay in VGPRs; NT hints keep
// the reused x0 table resident in the 192MB L2 while x1/out stream through.
// ---------------------------------------------------------------------------
__global__ __launch_bounds__(128) void
IWeightedSymmetricTensorProduct_kernel(const float* __restrict__ x0,
                                       const float* __restrict__ x1,
                                       const int* __restrict__ i0,
                                       const unsigned* __restrict__ ws,
                                       float* __restrict__ out) {
  // 4 waves * 2 buffers * 2048 floats = 64KB
  __shared__ __align__(16) float lds[4 * 4096];
  const int lane  = threadIdx.x & 31;     // u channel
  const int wave  = threadIdx.x >> 5;
  const int gwave = blockIdx.x * (blockDim.x >> 5) + wave;
  const int nwav  = gridDim.x * (blockDim.x >> 5);
  const int NPAIR = NB / 2;               // 25000

  float* wl = lds + wave * 4096;                       // wave-private region
  const unsigned wbase = (unsigned)(uintptr_t)wl;      // LDS byte offset
  const unsigned* rec = ws + WS_REC;

  // Stage one row pair (x1[2p],x1[2p+1] contiguous 4KB; g rows gathered)
  // into the LDS buffer at byte offset `base`: 16 async B128 ops.
  auto stage = [&](int p, unsigned base) {
    const int zi0 = __builtin_amdgcn_readfirstlane(i0[2 * p]);
    const int zi1 = __builtin_amdgcn_readfirstlane(i0[2 * p + 1]);
    const uint64_t gx1 = (uint64_t)(uintptr_t)(x1 + (size_t)(2 * p) * ROW);
    const uint64_t gg0 = (uint64_t)(uintptr_t)(x0 + (size_t)zi0 * ROW);
    const uint64_t gg1 = (uint64_t)(uintptr_t)(x0 + (size_t)zi1 * ROW);
#pragma unroll
    for (int q = 0; q < 8; ++q) {         // 4KB: both x1 rows, streamed (NT)
      unsigned la = base + q * 512 + lane * 16;
      uint64_t ga = gx1 + (unsigned)(q * 512 + lane * 16);
      asm volatile("global_load_async_to_lds_b128 %0, %1, off th:TH_LOAD_NT"
                   :: "v"(la), "v"(ga) : "memory");
    }
#pragma unroll
    for (int q = 0; q < 4; ++q) {         // g rows: reused table, keep RT
      unsigned la0 = base + 4096 + q * 512 + lane * 16;
      uint64_t ga0 = gg0 + (unsigned)(q * 512 + lane * 16);
      asm volatile("global_load_async_to_lds_b128 %0, %1, off"
                   :: "v"(la0), "v"(ga0) : "memory");
      unsigned la1 = base + 6144 + q * 512 + lane * 16;
      uint64_t ga1 = gg1 + (unsigned)(q * 512 + lane * 16);
      asm volatile("global_load_async_to_lds_b128 %0, %1, off"
                   :: "v"(la1), "v"(ga1) : "memory");
    }
  };

  if (gwave < NPAIR) stage(gwave, wbase);              // prime buffer 0

  unsigned buf = 0;
#pragma unroll 1
  for (int p = gwave; p < NPAIR; p += nwav) {
    int pn = p + nwav;
    if (pn >= NPAIR) pn = p;                           // clamped tail prefetch
    stage(pn, wbase + ((buf ^ 1) * 8192u));            // prefetch next pair
#if __has_builtin(__builtin_amdgcn_s_wait_asynccnt)
    __builtin_amdgcn_s_wait_asynccnt(16);              // current buffer landed
#else
    asm volatile("s_wait_asynccnt 16" ::: "memory");
#endif

    const float* X = wl + buf * 2048;   // X row0: [s*32+lane]; row1 at +512
    const float* G = X + 1024;          // G row0; row1 at +512
    float* o0 = out + (size_t)(2 * p) * ROW + lane;
    float* o1 = o0 + ROW;

#pragma unroll
    for (int s2 = 0; s2 < SDIM; ++s2) {
      float acc0 = 0.f, acc1 = 0.f;
      const unsigned j0 = ws[WS_OFF + s2 * 3 + 0];
      const unsigned j1 = ws[WS_OFF + s2 * 3 + 1];
      const unsigned j2 = ws[WS_OFF + s2 * 3 + 2];
      const unsigned j3 = ws[WS_OFF + s2 * 3 + 3];
      // degree 1
#pragma unroll 1
      for (unsigned j = j0; j < j1; ++j) {
        const unsigned pk = rec[2 * j];
        const float c = ((const float*)rec)[2 * j + 1];
        const int ig = (pk & 15) * UDIM + lane;
        const int ia = ((pk >> 4) & 15) * UDIM + lane;
        acc0 = fmaf(c * G[ig], X[ia], acc0);
        acc1 = fmaf(c * G[ig + 512], X[ia + 512], acc1);
      }
      // degree 2
#pragma unroll 1
      for (unsigned j = j1; j < j2; ++j) {
        const unsigned pk = rec[2 * j];
        const float c = ((const float*)rec)[2 * j + 1];
        const int ig = (pk & 15) * UDIM + lane;
        const int ia = ((pk >> 4) & 15) * UDIM + lane;
        const int ib = ((pk >> 8) & 15) * UDIM + lane;
        acc0 = fmaf(c * G[ig], X[ia] * X[ib], acc0);
        acc1 = fmaf(c * G[ig + 512], X[ia + 512] * X[ib + 512], acc1);
      }
      // degree 3
#pragma unroll 1
      for (unsigned j = j2; j < j3; ++j) {
        const unsigned pk = rec[2 * j];
        const float c = ((const float*)rec)[2 * j + 1];
        const int ig = (pk & 15) * UDIM + lane;
        const int ia = ((pk >> 4) & 15) * UDIM + lane;
        const int ib = ((pk >> 8) & 15) * UDIM + lane;
        const int ic = ((pk >> 12) & 15) * UDIM + lane;
        acc0 = fmaf(c * G[ig], X[ia] * X[ib] * X[ic], acc0);
        acc1 = fmaf(c * G[ig + 512], X[ia + 512] * X[ib + 512] * X[ic + 512],
                    acc1);
      }
      __builtin_nontemporal_store(acc0, &o0[s2 * UDIM]);  // streamed once
      __builtin_nontemporal_store(acc1, &o1[s2 * UDIM]);
    }
    buf ^= 1;
  }
}

extern "C" void kernel_launch(void* const* d_in, const int* in_sizes, int n_in,
                              void* d_out, int out_size, void* d_ws, size_t ws_size,
                              hipStream_t stream) {
  const float* x0   = (const float*)d_in[0];
  const float* x1   = (const float*)d_in[1];
  const float* c1   = (const float*)d_in[2];
  const float* c2   = (const float*)d_in[3];
  const float* c3   = (const float*)d_in[4];
  const int*   i0   = (const int*)d_in[5];
  const int*   p0_1 = (const int*)d_in[6];
  const int*   p1_1 = (const int*)d_in[7];
  const int*   p2_1 = (const int*)d_in[8];
  const int*   p0_2 = (const int*)d_in[9];
  const int*   p1_2 = (const int*)d_in[10];
  const int*   p2_2 = (const int*)d_in[11];
  const int*   p0_3 = (const int*)d_in[12];
  const int*   p1_3 = (const int*)d_in[13];
  const int*   p2_3 = (const int*)d_in[14];
  unsigned* ws = (unsigned*)d_ws;
  float* out = (float*)d_out;

  setup_paths<<<1, 32, 0, stream>>>(c1, c2, c3,
                                    p0_1, p1_1, p2_1,
                                    p0_2, p1_2, p2_2,
                                    p0_3, p1_3, p2_3, ws);
  // 2048 blocks * 4 waves = 8192 waves, ~3 row-pairs each.
  IWeightedSymmetricTensorProduct_kernel<<<2048, 128, 0, stream>>>(
      x0, x1, i0, ws, out);
}